// EdgeConv3d_5016521801768
// MI455X (gfx1250) — compile-verified
//
#include <hip/hip_runtime.h>

// EdgeConv restructured: W·[x_i ; x_j - x_i] + b == (W1-W2)·x_i + b  +  W2·x_j
//   Kernel 1: U[n] = (W1-W2)·x[n] + b ; V[n] = W2·x[n]   (WMMA f32 16x16x4 GEMM)
//   Kernel 2: out[b,:,n] = max_k relu(U[idx1[n,k]] + V[idx0[n,k]])  (L2 gather)

typedef __attribute__((ext_vector_type(2))) float v2f;
typedef __attribute__((ext_vector_type(4))) float v4f;
typedef __attribute__((ext_vector_type(8))) float v8f;

#define BB   2
#define CIN  64
#define COUT 64
#define NN   16384
#define KE   16

// ---------------------------------------------------------------------------
// Kernel 1: per-node tables via V_WMMA_F32_16X16X4_F32.
// One wave handles a 16-node tile; M = 64 output channels (4 WMMA M-tiles),
// K = 64 input channels (16 steps of 4). Two accumulator sets: U and V.
// ---------------------------------------------------------------------------
__global__ __launch_bounds__(256) void edgeconv_precompute(
    const float* __restrict__ x,      // [B, CIN, N]
    const float* __restrict__ W,      // [COUT, 2*CIN] row-major
    const float* __restrict__ bias,   // [COUT]
    float* __restrict__ U,            // [B*N, 64] node-major
    float* __restrict__ V)            // [B*N, 64] node-major
{
    const int lane = threadIdx.x & 31;
    const int wave = threadIdx.x >> 5;
    const int tile = blockIdx.x * 8 + wave;      // 0 .. B*N/16 - 1
    const int g0   = tile * 16;                  // global node base
    const int b    = g0 / NN;
    const int n0   = g0 - b * NN;

    const float* xb = x + (size_t)b * CIN * NN;

    v8f accU[4], accV[4];
#pragma unroll
    for (int m = 0; m < 4; ++m) { accU[m] = (v8f)0.f; accV[m] = (v8f)0.f; }

    const int nl = n0 + (lane & 15);   // node column this lane feeds (B-matrix N)
    const int hi = lane >> 4;          // 0: holds K0/K1, 1: holds K2/K3

    for (int kk = 0; kk < CIN; kk += 4) {
        const int c0 = kk + hi * 2;
        // B fragment: X tile 4x16 (K x N), f32 layout: VGPR0 = K0(/K2), VGPR1 = K1(/K3)
        v2f bf;
        bf.x = xb[(size_t)c0 * NN + nl];
        bf.y = xb[(size_t)(c0 + 1) * NN + nl];

#pragma unroll
        for (int m = 0; m < 4; ++m) {
            const int ch = m * 16 + (lane & 15);     // A-matrix row (M) per lane
            const float* wr = W + ch * (2 * CIN) + c0;
            const float w1a = wr[0],   w1b = wr[1];
            const float w2a = wr[CIN], w2b = wr[CIN + 1];
            v2f au; au.x = w1a - w2a; au.y = w1b - w2b;   // (W1 - W2) tile
            v2f av; av.x = w2a;       av.y = w2b;         // W2 tile
            accU[m] = __builtin_amdgcn_wmma_f32_16x16x4_f32(
                false, au, false, bf, (short)0, accU[m], false, false);
            accV[m] = __builtin_amdgcn_wmma_f32_16x16x4_f32(
                false, av, false, bf, (short)0, accV[m], false, false);
        }
    }

    // D layout: VGPR r -> M = r (lanes 0-15) / r+8 (lanes 16-31), N = lane&15.
    // Each lane thus holds 8 consecutive channels of one node: coalesced b128 stores.
    const int node = g0 + (lane & 15);
#pragma unroll
    for (int m = 0; m < 4; ++m) {
        const int chbase = m * 16 + hi * 8;
        float ub[8], vb[8];
#pragma unroll
        for (int r = 0; r < 8; ++r) {
            ub[r] = accU[m][r] + bias[chbase + r];
            vb[r] = accV[m][r];
        }
        float* up = U + (size_t)node * 64 + chbase;
        float* vp = V + (size_t)node * 64 + chbase;
        *(v4f*)(up)     = *(const v4f*)&ub[0];
        *(v4f*)(up + 4) = *(const v4f*)&ub[4];
        *(v4f*)(vp)     = *(const v4f*)&vb[0];
        *(v4f*)(vp + 4) = *(const v4f*)&vb[4];
    }
}

// ---------------------------------------------------------------------------
// Kernel 2: one wave per (b, n). Lanes 0-15 load the 16 neighbor (dir 0)
// indices, lanes 16-31 the 16 center (dir 1) indices; broadcast with shfl.
// Each lane owns channels {2*lane, 2*lane+1} -> float2 gathers are one
// coalesced 256B wave access per table per k (tables are L2-resident).
// relu + running max fold into fmaxf with acc initialized to 0.
// ---------------------------------------------------------------------------
__global__ __launch_bounds__(256) void edgeconv_gather_max(
    const long long* __restrict__ eidx,  // [2, B, N, K] int64
    const float* __restrict__ U,
    const float* __restrict__ V,
    float* __restrict__ out)             // [B, COUT, N]
{
    const int lane = threadIdx.x & 31;
    const int wave = threadIdx.x >> 5;
    const int gw   = blockIdx.x * 8 + wave;   // 0 .. B*N-1
    const int b    = gw / NN;
    const int n    = gw - b * NN;

    const int k   = lane & 15;
    const int dir = lane >> 4;                // 0 = neighbor (j), 1 = center (i)
    const size_t eoff = (((size_t)dir * BB + b) * NN + n) * KE + k;
    const int myidx = (int)eidx[eoff];

    const float* Ub = U + (size_t)b * NN * 64;
    const float* Vb = V + (size_t)b * NN * 64;

    float ax = 0.f, ay = 0.f;                 // relu(h) >= 0, so 0-init == relu+max
#pragma unroll
    for (int kk = 0; kk < KE; ++kk) {
        const int jj = __shfl(myidx, kk, 32);       // neighbor index (dir 0)
        const int ii = __shfl(myidx, 16 + kk, 32);  // center index   (dir 1)
        const float2 uu = ((const float2*)(Ub + (size_t)ii * 64))[lane];
        const float2 vv = ((const float2*)(Vb + (size_t)jj * 64))[lane];
        ax = fmaxf(ax, uu.x + vv.x);
        ay = fmaxf(ay, uu.y + vv.y);
    }

    out[((size_t)b * COUT + 2 * lane) * NN + n]     = ax;
    out[((size_t)b * COUT + 2 * lane + 1) * NN + n] = ay;
}

extern "C" void kernel_launch(void* const* d_in, const int* in_sizes, int n_in,
                              void* d_out, int out_size, void* d_ws, size_t ws_size,
                              hipStream_t stream) {
    const float*     x    = (const float*)d_in[0];
    const long long* eidx = (const long long*)d_in[1];
    const float*     W    = (const float*)d_in[2];
    const float*     bias = (const float*)d_in[3];

    float* U = (float*)d_ws;                       // B*N*64 floats = 8.4 MB
    float* V = U + (size_t)BB * NN * 64;           // another 8.4 MB
    float* out = (float*)d_out;

    // Kernel 1: B*N/16 node tiles, 8 waves (tiles) per 256-thread block.
    edgeconv_precompute<<<(BB * NN / 16) / 8, 256, 0, stream>>>(x, W, bias, U, V);
    // Kernel 2: one wave per (b,n), 8 waves per block.
    edgeconv_gather_max<<<(BB * NN) / 8, 256, 0, stream>>>(eidx, U, V, out);
}